// IpnLayer_11974368821316
// MI455X (gfx1250) — compile-verified
//
#include <hip/hip_runtime.h>

typedef __attribute__((ext_vector_type(2))) float v2f;
typedef __attribute__((ext_vector_type(8))) float v8f;

#define F        64
#define E        32
#define NPAIR    (F * (F - 1) / 2)   // 2016
#define LDSS     36                  // padded row stride (floats): 8B-aligned rows, bank-conflict-free fragment reads
#define NTILE    10                  // upper-triangular 16x16 tiles of the 4x4 tile grid
#define THREADS  (NTILE * 32)        // 10 waves, one tile per wave

__global__ __launch_bounds__(THREADS)
void gram_upper_tri_kernel(const float* __restrict__ x, float* __restrict__ out) {
    __shared__ float lds[F * LDSS];

    const int b     = blockIdx.x;
    const int t     = threadIdx.x;
    const float* xb = x + (size_t)b * (F * E);

    // ---- Stage batch (64x32 f32 = 8KB) into LDS, coalesced float2 loads ----
    for (int idx2 = t; idx2 < (F * E) / 2; idx2 += THREADS) {
        int row  = idx2 >> 4;       // E/2 = 16 float2 per row
        int col2 = idx2 & 15;
        v2f val = *(const v2f*)(xb + idx2 * 2);
        *(v2f*)(&lds[row * LDSS + col2 * 2]) = val;
    }
    __syncthreads();

    // ---- Wave -> tile mapping over upper triangle (incl. diagonal) of 4x4 ----
    const int w    = t >> 5;
    const int lane = t & 31;
    const int I = (w < 4) ? 0 : ((w < 7) ? 1 : ((w < 9) ? 2 : 3));
    const int s = (w < 4) ? 0 : ((w < 7) ? 4 : ((w < 9) ? 7 : 9));
    const int J = I + (w - s);

    // A (16x4 f32): lane<16 -> row M=lane, K={0,1}; lane>=16 -> row M=lane-16, K={2,3}
    // B (4x16 f32): mirror layout with N=lane&15 — since B = X^T, per-lane
    // addressing is identical to A, just a different row base.
    const int rsub = lane & 15;
    const int koff = (lane >> 4) << 1;
    const int arow = I * 16 + rsub;
    const int brow = J * 16 + rsub;

    v2f afrag[8], bfrag[8];
#pragma unroll
    for (int c = 0; c < 8; ++c) {
        afrag[c] = *(const v2f*)(&lds[arow * LDSS + c * 4 + koff]);
        bfrag[c] = *(const v2f*)(&lds[brow * LDSS + c * 4 + koff]);
    }

    // ---- 8 chained f32 WMMAs: K = 32 in steps of 4 ----
    v8f acc = {};
#pragma unroll
    for (int c = 0; c < 8; ++c) {
        acc = __builtin_amdgcn_wmma_f32_16x16x4_f32(
            /*neg_a=*/false, afrag[c],
            /*neg_b=*/false, bfrag[c],
            /*c_mod=*/(short)0, acc,
            /*reuse_a=*/false, /*reuse_b=*/false);
    }

    // ---- Scatter into packed strict-upper-triangle output ----
    // C/D layout: VGPR v, lanes 0-15: (M=v, N=lane); lanes 16-31: (M=v+8, N=lane-16)
    // Packed index p(i,j) = i*63 - i*(i-1)/2 + (j-i-1); recurrence p(i+1,j) = p(i,j) + (62-i).
    float* ob = out + (size_t)b * NPAIR;
    const int jg    = J * 16 + (lane & 15);
    const int ibase = I * 16 + ((lane >> 4) << 3);

    int p  = ibase * (F - 1) - (ibase * (ibase - 1)) / 2 + (jg - ibase - 1);
    int dp = (F - 2) - ibase;    // 62 - ibase

    if (I == J) {
        // Diagonal tile: per-element i<j mask (wave-uniform branch taken by 4/10 waves).
#pragma unroll
        for (int v = 0; v < 8; ++v) {
            if (ibase + v < jg) ob[p] = acc[v];
            p += dp; --dp;
        }
    } else {
        // Off-diagonal tile (I<J): every element is strictly upper, store unconditionally.
#pragma unroll
        for (int v = 0; v < 8; ++v) {
            ob[p] = acc[v];
            p += dp; --dp;
        }
    }
}

extern "C" void kernel_launch(void* const* d_in, const int* in_sizes, int n_in,
                              void* d_out, int out_size, void* d_ws, size_t ws_size,
                              hipStream_t stream) {
    (void)n_in; (void)out_size; (void)d_ws; (void)ws_size;
    const float* x = (const float*)d_in[0];
    float* out = (float*)d_out;
    const int nbatch = in_sizes[0] / (F * E);   // 8192 for the reference shapes
    gram_upper_tri_kernel<<<dim3(nbatch), dim3(THREADS), 0, stream>>>(x, out);
}